// MaskedBilinearFullSymLoss_78589311582543
// MI455X (gfx1250) — compile-verified
//
#include <hip/hip_runtime.h>
#include <hip/hip_bf16.h>
#include <stdint.h>

#define HW      512
#define NPIX    (HW * HW)
#define TW      64
#define TH      32
#define HALO    9
#define LW      (TW + HALO)        // 73 data columns
#define LSTRIDE (TW + HALO + 1)    // 74 padded stride
#define LROWS   (TH + HALO)        // 41 rows
#define PLANE   (LROWS * LSTRIDE)  // 3034 floats

typedef __attribute__((ext_vector_type(2))) float v2f;
typedef __attribute__((ext_vector_type(8))) float v8f;

typedef __attribute__((address_space(1))) int GInt;
typedef __attribute__((address_space(3))) int LInt;

__device__ __forceinline__ void async_copy_f32(const float* g, float* l) {
  GInt* gp = (GInt*)(uintptr_t)g;   // global addrspace, const dropped
  LInt* lp = (LInt*)l;              // generic -> LDS addrspace
#if __has_builtin(__builtin_amdgcn_global_load_async_to_lds_b32)
  __builtin_amdgcn_global_load_async_to_lds_b32(gp, lp, 0, 0);
#else
  asm volatile("global_load_async_to_lds_b32 %0, %1, off"
               :: "v"((uint32_t)(uintptr_t)lp),
                  "v"((unsigned long long)(uintptr_t)g)
               : "memory");
#endif
}

__device__ __forceinline__ void wait_async0() {
#if __has_builtin(__builtin_amdgcn_s_wait_asynccnt)
  __builtin_amdgcn_s_wait_asynccnt(0);
#else
  asm volatile("s_wait_asynccnt 0x0" ::: "memory");
#endif
}

// Exact f32 sum over the 32 lanes of a wave using V_WMMA_F32_16X16X4_F32.
// A holds each lane's value once (rest zeros), B = ones => D[m,n] = rowsum_A[m].
// D reg r, lanes 0..15 = rowsum[r]; lanes 16..31 = rowsum[r+8]. Sum 8 regs, fold halves.
__device__ __forceinline__ float wave_sum_wmma(float x) {
  v2f a; a[0] = x;    a[1] = 0.0f;
  v2f b; b[0] = 1.0f; b[1] = 1.0f;
  v8f c = {};
  v8f d = __builtin_amdgcn_wmma_f32_16x16x4_f32(false, a, false, b,
                                                (short)0, c, false, false);
  float s = ((d[0] + d[1]) + (d[2] + d[3])) + ((d[4] + d[5]) + (d[6] + d[7]));
  s += __shfl_xor(s, 16, 32);
  return s;
}

__global__ __launch_bounds__(256)
void loss_tile_kernel(const float* __restrict__ grid,
                      const float* __restrict__ gt,
                      const float* __restrict__ gd,
                      const float* __restrict__ mask,
                      float* __restrict__ ws, int nb) {
  __shared__ float L0[PLANE];
  __shared__ float L1[PLANE];
  __shared__ float LM[PLANE];
  __shared__ float red[8];

  const int tid  = threadIdx.x;
  const int bx   = blockIdx.x, by = blockIdx.y, bimg = blockIdx.z;
  const int c0   = bx * TW, r0 = by * TH;

  // Per-item scalars (uniform across the workgroup)
  const float dx = -8.0f * gt[bimg * 2 + 0];
  const float dy =  8.0f * gt[bimg * 2 + 1];
  const float sym_x = gd[bimg * 2 + 0];
  const float sym_y = gd[bimg * 2 + 1];
  const float dx1f = floorf(dx), dy1f = floorf(dy);
  const int   dx1 = (int)dx1f,  dy1 = (int)dy1f;
  const int   dx2 = dx1 + 1,    dy2 = dy1 + 1;
  const float wc1 = dx1f + 1.0f - dx, wc2 = dx - dx1f;
  const float wr1 = dy1f + 1.0f - dy, wr2 = dy - dy1f;
  const bool  pos = (dx > 0.0f);

  const float* g0 = grid + (size_t)(bimg * 2 + 0) * NPIX;
  const float* g1 = grid + (size_t)(bimg * 2 + 1) * NPIX;
  const float* mm = mask + (size_t)bimg * NPIX;

  // Stage tile + halo into LDS with clamped per-lane async gathers.
  // All stencil offsets are in [0,+9], so min(.,511) == reference clip.
  for (int t = tid; t < LROWS * LW; t += 256) {
    const int ly = t / LW;
    const int lx = t - ly * LW;
    const int gi = min(r0 + ly, HW - 1) * HW + min(c0 + lx, HW - 1);
    const int lo = ly * LSTRIDE + lx;
    async_copy_f32(g0 + gi, &L0[lo]);
    async_copy_f32(g1 + gi, &L1[lo]);
    async_copy_f32(mm + gi, &LM[lo]);
  }
  wait_async0();
  __syncthreads();

  float acc = 0.0f;
  const int tx  = tid & (TW - 1);
  const int ty0 = tid >> 6;          // 0..3
  const int j   = c0 + tx;

  if (pos) {
    const int limI = HW - dy2, limJ = HW - dx2;
#pragma unroll
    for (int k = 0; k < TH / 4; ++k) {
      const int ty   = ty0 + 4 * k;
      const int i    = r0 + ty;
      const int base = ty * LSTRIDE + tx;
      const int o11  = base + dy1 * LSTRIDE + dx1;
      const int o21  = base + dy2 * LSTRIDE + dx1;
      const float cv0 = L0[base], cv1 = L1[base];
      const float d0 = wc1 * wr1 * (cv0 - L0[o11]) + wc1 * wr2 * (cv0 - L0[o21])
                     + wc2 * wr1 * (cv0 - L0[o11 + 1]) + wc2 * wr2 * (cv0 - L0[o21 + 1]);
      const float d1 = wc1 * wr1 * (cv1 - L1[o11]) + wc1 * wr2 * (cv1 - L1[o21])
                     + wc2 * wr1 * (cv1 - L1[o11 + 1]) + wc2 * wr2 * (cv1 - L1[o21 + 1]);
      const float err = d0 * sym_y + d1 * sym_x;
      const float v   = (i < limI && j < limJ) ? LM[base] : 0.0f;
      acc = fmaf(v * err, err, acc);
    }
  } else {
    const int ndx1 = -dx1;                 // 0..8
    const int limI = HW - dy2, limJ = HW + dx1;
#pragma unroll
    for (int k = 0; k < TH / 4; ++k) {
      const int ty   = ty0 + 4 * k;
      const int i    = r0 + ty;
      const int base = ty * LSTRIDE + tx;
      const int oA   = base + dy2 * LSTRIDE;     // (i+dy2, j)
      const int oC   = base + LSTRIDE + ndx1;    // (i+1,  j-dx1)
      const int oD   = base + ndx1;              // (i,    j-dx1)
      const float A0 = L0[oA], B0 = L0[oA + 1], C0 = L0[oC], D0 = L0[oD];
      const float A1 = L1[oA], B1 = L1[oA + 1], C1 = L1[oC], D1 = L1[oD];
      const float d0 = wc1 * wr1 * (A0 - C0) + wc1 * wr2 * (A0 - D0)
                     + wc2 * wr1 * (B0 - C0) + wc2 * wr2 * (B0 - D0);
      const float d1 = wc1 * wr1 * (A1 - C1) + wc1 * wr2 * (A1 - D1)
                     + wc2 * wr1 * (B1 - C1) + wc2 * wr2 * (B1 - D1);
      const float err = d0 * sym_y + d1 * sym_x;
      const float mN  = LM[oA + ndx1];           // m at (i+dy2, j-dx1)
      const float v   = (i < limI && j < limJ) ? mN : 0.0f;
      acc = fmaf(v * err, err, acc);
    }
  }

  // Wave sum on the matrix pipe, then 8 waves -> 1 value.
  const float s = wave_sum_wmma(acc);
  if ((tid & 31) == 0) red[tid >> 5] = s;
  __syncthreads();
  if (tid == 0) {
    float tot = 0.0f;
#pragma unroll
    for (int w = 0; w < 8; ++w) tot += red[w];
    const float cnt = pos ? (float)(HW - dy2) * (float)(HW - dx2)
                          : (float)(HW - dy2) * (float)(HW + dx1);
    const int wgid = (bimg * (int)gridDim.y + by) * (int)gridDim.x + bx;
    ws[wgid] = tot / (cnt * (float)nb);
  }
}

__global__ __launch_bounds__(256)
void final_reduce_kernel(const float* __restrict__ ws, float* __restrict__ out, int n) {
  __shared__ float sh[256];
  float s = 0.0f;
  for (int i = threadIdx.x; i < n; i += 256) s += ws[i];
  sh[threadIdx.x] = s;
  __syncthreads();
  for (int off = 128; off > 0; off >>= 1) {
    if ((int)threadIdx.x < off) sh[threadIdx.x] += sh[threadIdx.x + off];
    __syncthreads();
  }
  if (threadIdx.x == 0) out[0] = sh[0];
}

extern "C" void kernel_launch(void* const* d_in, const int* in_sizes, int n_in,
                              void* d_out, int out_size, void* d_ws, size_t ws_size,
                              hipStream_t stream) {
  (void)n_in; (void)out_size; (void)ws_size;
  const float* grid = (const float*)d_in[0];
  const float* gt   = (const float*)d_in[1];
  const float* gd   = (const float*)d_in[2];
  const float* mask = (const float*)d_in[3];
  float* out = (float*)d_out;
  float* ws  = (float*)d_ws;

  const int nb = in_sizes[1] / 2;              // batch size (16)
  dim3 grd(HW / TW, HW / TH, nb);              // 8 x 16 x 16 = 2048 workgroups
  loss_tile_kernel<<<grd, 256, 0, stream>>>(grid, gt, gd, mask, ws, nb);

  const int n = (HW / TW) * (HW / TH) * nb;
  final_reduce_kernel<<<1, 256, 0, stream>>>(ws, out, n);
}